// MinkowskiUNetBackbone_33870112096420
// MI455X (gfx1250) — compile-verified
//
#include <hip/hip_runtime.h>

// Sparse conv backbone: h = relu(spconv(x, W_in)); out = spconv(h, W_out)
// Gather (async global->LDS) -> fp32 WMMA GEMM -> atomic scatter-add,
// per (offset k, pair-tile).

#define K_OFF 27
#define CIN1  32
#define CMID  64
#define COUT2 32
#define TP    128      // pairs per block
#define NTHREADS 256   // 8 waves (wave32)

typedef __attribute__((ext_vector_type(2))) float v2f;
typedef __attribute__((ext_vector_type(8))) float v8f;

__global__ __launch_bounds__(NTHREADS)
void zero2_kernel(float4* __restrict__ a, long n4a, float4* __restrict__ b, long n4b) {
    const long stride = (long)gridDim.x * blockDim.x;
    const float4 z = make_float4(0.f, 0.f, 0.f, 0.f);
    for (long i = (long)blockIdx.x * blockDim.x + threadIdx.x; i < n4a; i += stride) a[i] = z;
    for (long i = (long)blockIdx.x * blockDim.x + threadIdx.x; i < n4b; i += stride) b[i] = z;
}

__global__ __launch_bounds__(NTHREADS)
void relu_kernel(float* __restrict__ h, long n) {
    const long stride = (long)gridDim.x * blockDim.x;
    for (long i = (long)blockIdx.x * blockDim.x + threadIdx.x; i < n; i += stride)
        h[i] = fmaxf(h[i], 0.0f);
}

// One block: offset k = blockIdx.y, pair rows [blockIdx.x*TP, ...+TP).
// Gathered rows are DMA'd straight into LDS with global_load_async_to_lds_b128;
// mask is applied post-GEMM at scatter time ((m*x)W == m*(xW)).
template <int CIN, int COUT>
__global__ __launch_bounds__(NTHREADS)
void spconv_kernel(const float* __restrict__ x,
                   const int*   __restrict__ in_idx,
                   const int*   __restrict__ out_idx,
                   const float* __restrict__ mask,
                   const float* __restrict__ Wall,
                   float*       __restrict__ out,
                   int P) {
    __shared__ int    sIn[TP];
    __shared__ int    sOut[TP];
    __shared__ float  sM[TP];
    __shared__ float  sX[TP][CIN];              // gathered features (async DMA dest)
    __shared__ float2 sW[(CIN / 2) * COUT];     // sW[kp*COUT+n] = {W[2kp][n], W[2kp+1][n]}

    const int tid = threadIdx.x;
    const int k  = blockIdx.y;
    const int p0 = blockIdx.x * TP;
    const float* __restrict__ W = Wall + (long)k * CIN * COUT;

    // Stage pair indices + validity mask.
    for (int r = tid; r < TP; r += NTHREADS) {
        const int p = p0 + r;
        const bool ok = (p < P);
        const long ofs = (long)k * P + p;
        sIn[r]  = ok ? in_idx[ofs]  : 0;
        sOut[r] = ok ? out_idx[ofs] : 0;
        sM[r]   = (ok && mask[ofs] > 0.5f) ? 1.0f : 0.0f;
    }
    // Stage W[k] with K-row pairs interleaved (one ds_load_b64 per B fragment).
    for (int i = tid; i < (CIN / 2) * COUT; i += NTHREADS) {
        const int kp = i / COUT;
        const int n  = i - kp * COUT;
        sW[i] = make_float2(W[(2 * kp) * COUT + n], W[(2 * kp + 1) * COUT + n]);
    }
    __syncthreads();   // sIn visible to all threads before gather

    // Async gather: 16B chunks of feature rows DMA'd global -> LDS.
    // LDS dest operand = low 32 bits of the flat shared address.
    constexpr int C4 = CIN / 4;
    for (int i = tid; i < TP * C4; i += NTHREADS) {
        const int row = i / C4;
        const int c4  = i - row * C4;
        const float* gsrc = x + (long)sIn[row] * CIN + c4 * 4;
        const unsigned ldst = (unsigned)(unsigned long long)(&sX[row][c4 * 4]);
        asm volatile("global_load_async_to_lds_b128 %0, %1, off"
                     :: "v"(ldst), "v"(gsrc) : "memory");
    }
    // Wait for this wave's async DMA before the barrier (asm not compiler-tracked).
    asm volatile("s_wait_asynccnt 0x0" ::: "memory");
    __syncthreads();

    // WMMA: each wave owns M-tile = waveId, loops over N-tiles.
    const int wave = tid >> 5;
    const int lane = tid & 31;
    const int ll = lane & 15;        // lane within half-wave
    const int hi = lane >> 4;        // 0: lanes 0-15, 1: lanes 16-31
    constexpr int NT = COUT / 16;

    for (int t = wave; t < 8 * NT; t += NTHREADS / 32) {
        const int m0 = (t & 7) * 16;
        const int n0 = (t >> 3) * 16;
        v8f c = {0.f, 0.f, 0.f, 0.f, 0.f, 0.f, 0.f, 0.f};
#pragma unroll
        for (int kk = 0; kk < CIN / 4; ++kk) {
            // A 16x4 f32: lanes 0-15 hold K{4kk,4kk+1}; lanes 16-31 K{4kk+2,4kk+3}
            v2f a = *((const v2f*)&sX[m0 + ll][4 * kk + 2 * hi]);
            // B 4x16 f32: symmetric layout, column = n0+ll
            const float2 bw = sW[(2 * kk + hi) * COUT + (n0 + ll)];
            v2f b; b[0] = bw.x; b[1] = bw.y;
            c = __builtin_amdgcn_wmma_f32_16x16x4_f32(
                    false, a, false, b, (short)0, c, false, false);
        }
        // Scatter: VGPR r holds D[M = r + 8*hi][N = ll]; gate by mask here.
#pragma unroll
        for (int r = 0; r < 8; ++r) {
            const int  prow = m0 + r + 8 * hi;
            const long orow = sOut[prow];
            atomicAdd(out + orow * COUT + (n0 + ll), c[r] * sM[prow]);
        }
    }
}

extern "C" void kernel_launch(void* const* d_in, const int* in_sizes, int n_in,
                              void* d_out, int out_size, void* d_ws, size_t ws_size,
                              hipStream_t stream) {
    const float* feats   = (const float*)d_in[0];
    const int*   in_idx  = (const int*)d_in[1];
    const int*   out_idx = (const int*)d_in[2];
    const float* mask    = (const float*)d_in[3];
    const float* W_in    = (const float*)d_in[4];
    const float* W_out   = (const float*)d_in[5];
    float* out = (float*)d_out;
    float* h   = (float*)d_ws;   // intermediate [N, CMID] fp32

    const int N = in_sizes[0] / CIN1;
    const int P = in_sizes[3] / K_OFF;

    const long nh = (long)N * CMID;
    const long no = (long)N * COUT2;

    // Zero accumulators every call (harness does not re-poison between replays).
    zero2_kernel<<<2048, NTHREADS, 0, stream>>>((float4*)h, nh / 4, (float4*)out, no / 4);

    dim3 grid((P + TP - 1) / TP, K_OFF);
    spconv_kernel<CIN1, CMID><<<grid, NTHREADS, 0, stream>>>(
        feats, in_idx, out_idx, mask, W_in, h, P);
    relu_kernel<<<2048, NTHREADS, 0, stream>>>(h, nh);
    spconv_kernel<CMID, COUT2><<<grid, NTHREADS, 0, stream>>>(
        h, in_idx, out_idx, mask, W_out, out, P);
}